// PointTriplaneGenerator_15144054686228
// MI455X (gfx1250) — compile-verified
//
#include <hip/hip_runtime.h>
#include <hip/hip_bf16.h>
#include <cstddef>

// ---------------------------------------------------------------------------
// PointTriplaneGenerator for MI455X (gfx1250, wave32)
//
// Phase 1: zero 77MB plane accumulator in d_ws            (stream stores)
// Phase 2: Gaussian splat, one wave32 per point           (global_atomic_add_f32,
//          32 contiguous channels per atomic burst -> 128B contiguous, L2-resident)
// Phase 3: 1x1 conv as GEMM  out[o,pix] = sum_c W[o,c]*P[pix,c] + b[o]
//          via V_WMMA_F32_16X16X4_F32 (exact f32 precision; the GEMM is
//          bandwidth-bound at 23.3 TB/s so low-precision WMMA buys nothing).
//          12 of 13 M-tiles take a fully unconditional inner loop; only the
//          partial tile (o=192..207) uses a branchless clamp+mask.
// ---------------------------------------------------------------------------

#define G 128
#define CF 196
#define NPTS 100000
#define NB 2
#define PIX (G * G)                 // 16384 pixels per plane
#define NPLANES (NB * 3)            // 6 (b,plane) pairs
#define OPACITY 0.0067f
#define INV2S2 2.0f                 // 0.5 / (0.5*0.5)

typedef __attribute__((ext_vector_type(2))) float v2f;
typedef __attribute__((ext_vector_type(8))) float v8f;

__device__ __forceinline__ void atomic_add_f32(float* p, float v) {
    // relaxed, device-scope -> global_atomic_add_f32 on gfx1250
    __hip_atomic_fetch_add(p, v, __ATOMIC_RELAXED, __HIP_MEMORY_SCOPE_AGENT);
}

// --------------------------- Phase 1: zero workspace -----------------------
__global__ __launch_bounds__(256) void zero_kernel(float4* __restrict__ p, size_t n4) {
    size_t i = (size_t)blockIdx.x * blockDim.x + threadIdx.x;
    size_t stride = (size_t)gridDim.x * blockDim.x;
    float4 z = make_float4(0.f, 0.f, 0.f, 0.f);
    for (; i < n4; i += stride) p[i] = z;
}

// --------------------------- Phase 2: splat --------------------------------
// One wave32 per point. Point-uniform work (coords, tap weights) is computed
// redundantly per lane; the 196 channels are strided across lanes so each
// atomic burst covers 32 consecutive floats of one pixel's channel vector.
__global__ __launch_bounds__(256) void splat_kernel(
    const float* __restrict__ dense,   // [B, N, 199] (x,y,z, 196 feats)
    const float* __restrict__ weight,  // [B, N]
    const float* __restrict__ sb,      // [6] scene bounds
    float* __restrict__ planes)        // [NPLANES, PIX, CF]
{
    const int wave  = threadIdx.x >> 5;
    const int lane  = threadIdx.x & 31;
    const int point = blockIdx.x * 8 + wave;
    if (point >= NB * NPTS) return;
    const int b = point / NPTS;

    const float* pt  = dense + (size_t)point * 199;
    const float  wgt = weight[point];

    const float nx = 2.f * (pt[0] - sb[0]) / (sb[1] - sb[0]) - 1.f;
    const float ny = 2.f * (pt[1] - sb[2]) / (sb[3] - sb[2]) - 1.f;
    const float nz = 2.f * (pt[2] - sb[4]) / (sb[5] - sb[4]) - 1.f;

    float u[3];
    u[0] = (nx + 1.f) * 0.5f * (float)G - 0.5f;
    u[1] = (ny + 1.f) * 0.5f * (float)G - 0.5f;
    u[2] = (nz + 1.f) * 0.5f * (float)G - 0.5f;

    // colors = feats * weight, strided over lanes: chunk j holds channel lane+32j
    float col[7];
#pragma unroll
    for (int j = 0; j < 7; ++j) {
        int c = lane + 32 * j;
        col[j] = (c < CF) ? pt[3 + c] * wgt : 0.f;   // folds for j<6 (lane+160<196)
    }

    const int axA[3] = {0, 0, 1};   // xy, xz, yz
    const int axB[3] = {1, 2, 2};

#pragma unroll
    for (int p = 0; p < 3; ++p) {
        const float ua = u[axA[p]];          // -> ix (W)
        const float ub = u[axB[p]];          // -> iy (H)
        const int ia = (int)floorf(ua);
        const int ib = (int)floorf(ub);
        float* base = planes + (size_t)(b * 3 + p) * PIX * CF;
#pragma unroll
        for (int dy = -1; dy <= 1; ++dy) {
#pragma unroll
            for (int dx = -1; dx <= 1; ++dx) {
                const int jx = ia + dx;
                const int jy = ib + dy;
                if ((unsigned)jx < (unsigned)G && (unsigned)jy < (unsigned)G) {
                    const float ddx = (float)jx - ua;
                    const float ddy = (float)jy - ub;
                    const float w = OPACITY * __expf(-(ddx * ddx + ddy * ddy) * INV2S2);
                    float* dst = base + (size_t)(jy * G + jx) * CF;
#pragma unroll
                    for (int j = 0; j < 7; ++j) {
                        int c = lane + 32 * j;
                        if (c < CF) atomic_add_f32(dst + c, w * col[j]);
                    }
                }
            }
        }
    }
}

// --------------------------- Phase 3: WMMA GEMM ----------------------------
// out[bp][o][pix] = sum_c conv_w[o][c] * planes[bp][pix][c] + conv_b[o]
// M = o (196 -> 13 tiles of 16), N = pix (16 per block), K = c (49 tiles of 4).
//
// ISA f32 fragment layouts (wave32, CDNA5 ISA §7.12.2):
//   A 16x4:  lane l holds A[M = l&15][K = 2*(l>>4) + {0,1}] in {v0,v1}
//   B 4x16:  lane l holds B[K = 2*(l>>4) + {0,1}][N = l&15] in {v0,v1}
//   C/D 16x16: vgpr r, lane l -> element (M = r + 8*(l>>4), N = l&15)
template <bool PARTIAL>
__device__ __forceinline__ void conv_tile(
    const float* __restrict__ conv_w, const float* __restrict__ conv_b,
    const float* __restrict__ lds_b, float* __restrict__ obase,
    int m0, int pix0, int lhalf, int l15)
{
    const int  arow   = m0 + l15;                     // output-channel row of A
    float      avalid = 1.f;
    int        arow_c = arow;
    if (PARTIAL) {                                    // branchless clamp + mask
        avalid = (arow < CF) ? 1.f : 0.f;
        arow_c = (arow < CF) ? arow : (CF - 1);
    }
    // row*196 + 2*lhalf is even -> 8B-aligned v2f (b64) loads
    const float* aptr = conv_w + (size_t)arow_c * CF + 2 * lhalf;
    const float* bptr = lds_b + l15 * CF + 2 * lhalf;

    v8f acc = {};
#pragma unroll 7
    for (int k0 = 0; k0 < CF; k0 += 4) {              // 49 = 7x7 K-steps
        v2f a  = *(const v2f*)(aptr + k0);
        v2f bf = *(const v2f*)(bptr + k0);
        if (PARTIAL) a *= avalid;
        acc = __builtin_amdgcn_wmma_f32_16x16x4_f32(
            false, a, false, bf, (short)0, acc, false, false);
    }

    // Store: 16 consecutive pixels per half-wave per register -> 64B coalesced;
    // r*PIX folds into the 24-bit instruction offset (max 448KB).
    float* optr = obase + (size_t)(m0 + 8 * lhalf) * PIX + pix0 + l15;
    if (!PARTIAL) {
        const float* bias = conv_b + m0 + 8 * lhalf;  // max index 191 < 196
#pragma unroll
        for (int r = 0; r < 8; ++r)
            optr[(size_t)r * PIX] = acc[r] + bias[r];
    } else {
#pragma unroll
        for (int r = 0; r < 8; ++r) {
            const int o = m0 + 8 * lhalf + r;
            if (o < CF) optr[(size_t)r * PIX] = acc[r] + conv_b[o];
        }
    }
}

__global__ __launch_bounds__(256) void conv_kernel(
    const float* __restrict__ planes,  // [NPLANES, PIX, CF]
    const float* __restrict__ conv_w,  // [CF, CF] row-major (o, c)
    const float* __restrict__ conv_b,  // [CF]
    float* __restrict__ out)           // [NPLANES, CF, PIX]
{
    __shared__ float lds_b[16 * CF];   // 16 pixels x 196 channels (12.25 KB)

    const int tile = blockIdx.x;        // NPLANES * (PIX/16) = 6144 tiles
    const int bp   = tile >> 10;        // 1024 pixel-tiles per plane
    const int pix0 = (tile & 1023) * 16;

    // Stage the 16-pixel B-tile once; reused by all 13 M-tiles.
    const float* bsrc = planes + ((size_t)bp * PIX + pix0) * CF;
    for (int i = threadIdx.x; i < 16 * CF; i += 256) lds_b[i] = bsrc[i];
    __syncthreads();

    const int wave  = threadIdx.x >> 5;
    const int lhalf = (threadIdx.x >> 4) & 1;
    const int l15   = threadIdx.x & 15;

    float* obase = out + (size_t)bp * CF * PIX;

    // 13 M-tiles over 8 waves: tiles 0..7 (full), 8..11 (full), 12 (partial).
    conv_tile<false>(conv_w, conv_b, lds_b, obase, wave * 16, pix0, lhalf, l15);
    if (wave < 4)
        conv_tile<false>(conv_w, conv_b, lds_b, obase, (8 + wave) * 16, pix0, lhalf, l15);
    else if (wave == 4)
        conv_tile<true>(conv_w, conv_b, lds_b, obase, 12 * 16, pix0, lhalf, l15);
}

// ---------------------------------------------------------------------------
extern "C" void kernel_launch(void* const* d_in, const int* in_sizes, int n_in,
                              void* d_out, int out_size, void* d_ws, size_t ws_size,
                              hipStream_t stream) {
    (void)in_sizes; (void)n_in; (void)out_size; (void)ws_size;
    // setup_inputs order: central(unused), dense, weight, scene_bounds, conv_w, conv_b
    const float* dense  = (const float*)d_in[1];
    const float* weight = (const float*)d_in[2];
    const float* sb     = (const float*)d_in[3];
    const float* conv_w = (const float*)d_in[4];
    const float* conv_b = (const float*)d_in[5];
    float* out    = (float*)d_out;
    float* planes = (float*)d_ws;   // needs NPLANES*PIX*CF*4 = 77,070,336 B

    const size_t n4 = (size_t)NPLANES * PIX * CF / 4;   // 4,816,896 float4
    zero_kernel<<<2048, 256, 0, stream>>>((float4*)planes, n4);

    const int npoints = NB * NPTS;
    splat_kernel<<<(npoints + 7) / 8, 256, 0, stream>>>(dense, weight, sb, planes);

    conv_kernel<<<NPLANES * (PIX / 16), 256, 0, stream>>>(planes, conv_w, conv_b, out);
}